// VAE_70952859730647
// MI455X (gfx1250) — compile-verified
//
#include <hip/hip_runtime.h>

#define LATENT  10
#define MAXLEN  80
#define BATCH   512
#define NSTEPS  100
#define HID     64
#define DT      0.01f
#define SQRT_DT 0.1f

typedef __attribute__((ext_vector_type(2))) float v2f;
typedef __attribute__((ext_vector_type(8))) float v8f;

#define XPAD 12            // X tile row stride (10 latent + t + zero pad)
#define XDUMP (16 * XPAD)  // dump zone for lanes 12..15 (branch-free scatter)
#define HPAD 68            // h tile row stride: 64 + 4 -> conflict-free on 64 banks

// One wave (32 lanes) owns a 16-token tile: fixed position l, 16 consecutive batch rows.
// WMMA f32 16x16x4 layouts (wave32):
//   A (16x4):  lane L -> M = L%16, K = r + 2*(L/16)      (r = vgpr 0..1)
//   B (4x16):  lane L -> N = L%16, K = r + 2*(L/16)
//   C/D(16x16):lane L -> N = L%16, M = r + 8*(L/16)      (r = vgpr 0..7)
__global__ __launch_bounds__(32, 1)
void sde_tile_kernel(const float* __restrict__ z_mean,
                     const float* __restrict__ z_log_var,
                     const float* __restrict__ W1,
                     const float* __restrict__ b1,
                     const float* __restrict__ W2,
                     const float* __restrict__ b2,
                     const float* __restrict__ noise,
                     float* __restrict__ out)
{
    __shared__ float ldsXT[16 * XPAD + 32];   // +32-float dump zone
    __shared__ float ldsH [16 * HPAD];

    const int lane   = threadIdx.x;       // 0..31
    const int half   = lane >> 4;         // 0/1
    const int lane16 = lane & 15;

    const int tile = blockIdx.x;          // 0..2559
    const int l    = tile >> 5;           // position 0..79
    const int b0   = (tile & 31) << 4;    // batch base (16 rows)

    // ---- weight fragments, B-layout, register resident (zero-padded) ----
    v2f w1f[3][4];
    #pragma unroll
    for (int kk = 0; kk < 3; ++kk)
        #pragma unroll
        for (int nn = 0; nn < 4; ++nn)
            #pragma unroll
            for (int r = 0; r < 2; ++r) {
                const int k = 4*kk + 2*half + r;
                const int n = 16*nn + lane16;
                w1f[kk][nn][r] = (k < LATENT + 1) ? W1[k*HID + n] : 0.0f;
            }
    v2f w2f[16];
    #pragma unroll
    for (int kk = 0; kk < 16; ++kk)
        #pragma unroll
        for (int r = 0; r < 2; ++r) {
            const int k = 4*kk + 2*half + r;
            w2f[kk][r] = (lane16 < LATENT) ? W2[k*LATENT + lane16] : 0.0f;
        }
    float b1v[4];
    #pragma unroll
    for (int nn = 0; nn < 4; ++nn) b1v[nn] = b1[16*nn + lane16];
    const float b2v = (lane16 < LATENT) ? b2[lane16] : 0.0f;

    // ---- per-element SDE state, C/D layout: elem r -> (row m = r+8*half, col lane16) ----
    // precomputed per-element constants: rsig = 1/sigma, hsv = 0.5*dt*sigma_var,
    // nstd = sqrt_dt*sigma_std  -> per-step update is pure FMA, no division.
    const bool act   = (lane16 < LATENT);
    const bool act12 = (lane16 < XPAD);
    v8f mu_d, rsig_d, hsv_d, nstd_d, xt_d, err_d;
    #pragma unroll
    for (int r = 0; r < 8; ++r) {
        const int m  = r + 8*half;
        const int bi = b0 + m;
        float mu = 0.f, rs = 1.f, hv = 0.5f * DT, ns = SQRT_DT;
        if (act) {
            const size_t base = ((size_t)bi * MAXLEN + l) * LATENT + lane16;
            const float zm  = z_mean[base];
            const float zlv = z_log_var[base];
            float zmp = 0.f, zlp = 0.f;
            if (l > 0) {
                const size_t bp = ((size_t)bi * MAXLEN + (l - 1)) * LATENT + lane16;
                zmp = z_mean[bp];
                zlp = z_log_var[bp];
            }
            mu = zm - zmp;                                     // shifted delta
            const float mx = fmaxf(zlv, zlp);
            const float sg = mx + log1pf(expf(-fabsf(zlv - zlp)));   // logaddexp
            const float st = expf(0.5f * sg);                  // sigma_std
            rs = 1.0f / sg;
            hv = 0.5f * DT * st * st;                          // 0.5*dt*sigma_var
            ns = SQRT_DT * st;                                 // sqrt_dt*sigma_std
        }
        mu_d[r] = mu; rsig_d[r] = rs; hsv_d[r] = hv; nstd_d[r] = ns;
        xt_d[r] = mu; err_d[r] = 0.f;
    }

    // branch-free scatter addresses: real slot for lane16<12, dump slot otherwise
    int xoff[8];
    #pragma unroll
    for (int r = 0; r < 8; ++r) {
        const int m = r + 8*half;
        xoff[r] = act12 ? (m * XPAD + lane16) : (XDUMP + lane);
    }
    // clamped noise column (inactive lanes read col 0, value discarded)
    const int ncol = act ? lane16 : 0;

    const float* nbase = noise + (size_t)l * NSTEPS * BATCH * LATENT;

    #pragma unroll 1
    for (int s = 0; s < NSTEPS; ++s) {
        const float t_s  = (float)(l + s) * DT;
        const float tpad = (lane16 == LATENT) ? t_s : 0.0f;   // col 10 = t, cols 11..15 = 0

        // (a) branch-free scatter of X tile (value + address both cndmask-selected)
        #pragma unroll
        for (int r = 0; r < 8; ++r)
            ldsXT[xoff[r]] = act ? xt_d[r] : tpad;
        __syncthreads();

        // prefetch next step's noise cacheline (address clamped to stay in-bounds)
        {
            const int sp = (s + 1 < NSTEPS) ? (s + 1) : s;
            __builtin_prefetch(nbase + (size_t)sp * BATCH * LATENT
                                     + (size_t)(b0 + 8*half) * LATENT, 0, 0);
        }

        // (b) gather X A-fragments (ds_load_2addr_b64, conflict-free)
        v2f xa[3];
        #pragma unroll
        for (int kk = 0; kk < 3; ++kk)
            xa[kk] = *(const v2f*)&ldsXT[lane16*XPAD + 4*kk + 2*half];

        // (c) GEMM1: h = X @ W1. Zero-init C -> inline-0 operand, 4 independent
        //     accumulator blocks, kk-major interleave for WMMA ILP.
        v8f acc[4] = {{}, {}, {}, {}};
        #pragma unroll
        for (int kk = 0; kk < 3; ++kk)
            #pragma unroll
            for (int nn = 0; nn < 4; ++nn)
                acc[nn] = __builtin_amdgcn_wmma_f32_16x16x4_f32(
                              false, xa[kk], false, w1f[kk][nn],
                              (short)0, acc[nn], false, false);

        // (d,e) bias + ReLU in D layout (dual-issued add/max), store h tile
        #pragma unroll
        for (int nn = 0; nn < 4; ++nn)
            #pragma unroll
            for (int r = 0; r < 8; ++r) {
                const int m = r + 8*half;
                ldsH[m*HPAD + 16*nn + lane16] = fmaxf(acc[nn][r] + b1v[nn], 0.0f);
            }
        __syncthreads();

        // (f) GEMM2: score = h @ W2, 4 round-robin accumulators (short RAW chains)
        v8f s0 = {}, s1 = {}, s2 = {}, s3 = {};
        #pragma unroll
        for (int kk = 0; kk < 16; kk += 4) {
            v2f ha0 = *(const v2f*)&ldsH[lane16*HPAD + 4*(kk+0) + 2*half];
            v2f ha1 = *(const v2f*)&ldsH[lane16*HPAD + 4*(kk+1) + 2*half];
            v2f ha2 = *(const v2f*)&ldsH[lane16*HPAD + 4*(kk+2) + 2*half];
            v2f ha3 = *(const v2f*)&ldsH[lane16*HPAD + 4*(kk+3) + 2*half];
            s0 = __builtin_amdgcn_wmma_f32_16x16x4_f32(false, ha0, false, w2f[kk+0],
                                                       (short)0, s0, false, false);
            s1 = __builtin_amdgcn_wmma_f32_16x16x4_f32(false, ha1, false, w2f[kk+1],
                                                       (short)0, s1, false, false);
            s2 = __builtin_amdgcn_wmma_f32_16x16x4_f32(false, ha2, false, w2f[kk+2],
                                                       (short)0, s2, false, false);
            s3 = __builtin_amdgcn_wmma_f32_16x16x4_f32(false, ha3, false, w2f[kk+3],
                                                       (short)0, s3, false, false);
        }
        const v8f sacc = (s0 + s1) + (s2 + s3);

        // (g) branch-free elementwise SDE update: pure FMA form
        //     xt' = fma(nstd, dw, fma(hsv, score, fma(mu, DT, xt)))
        //     df  = (mu - xt)*rsig - score ; err += df*df
        {
            const float* nz = nbase + (size_t)s * BATCH * LATENT;
            #pragma unroll
            for (int r = 0; r < 8; ++r) {
                const int m = r + 8*half;
                const float score = sacc[r] + b2v;
                const float df = (mu_d[r] - xt_d[r]) * rsig_d[r] - score;
                err_d[r] = fmaf(df, df, err_d[r]);
                const float dw = nz[(size_t)(b0 + m) * LATENT + ncol];
                float x = fmaf(mu_d[r], DT, xt_d[r]);
                x = fmaf(hsv_d[r], score, x);
                xt_d[r] = fmaf(nstd_d[r], dw, x);
            }
        }
        __syncthreads();   // protect ldsXT/ldsH reuse next iteration
    }

    // ---- outputs: out[2, B, L, D] = {sequence, mean score_error} ----
    if (act) {
        #pragma unroll
        for (int r = 0; r < 8; ++r) {
            const int m  = r + 8*half;
            const int bi = b0 + m;
            const size_t o0 = ((size_t)bi * MAXLEN + l) * LATENT + lane16;
            const size_t o1 = ((size_t)(BATCH + bi) * MAXLEN + l) * LATENT + lane16;
            out[o0] = xt_d[r];
            out[o1] = err_d[r] * (1.0f / NSTEPS);
        }
    }
}

extern "C" void kernel_launch(void* const* d_in, const int* in_sizes, int n_in,
                              void* d_out, int out_size, void* d_ws, size_t ws_size,
                              hipStream_t stream) {
    (void)in_sizes; (void)n_in; (void)out_size; (void)d_ws; (void)ws_size;
    const float* z_mean    = (const float*)d_in[0];
    const float* z_log_var = (const float*)d_in[1];
    const float* W1        = (const float*)d_in[2];
    const float* b1        = (const float*)d_in[3];
    const float* W2        = (const float*)d_in[4];
    const float* b2        = (const float*)d_in[5];
    const float* noise     = (const float*)d_in[6];
    float* out             = (float*)d_out;

    const int numTiles = (MAXLEN * BATCH) / 16;   // 2560 tiles, 1 wave each
    sde_tile_kernel<<<dim3(numTiles), dim3(32), 0, stream>>>(
        z_mean, z_log_var, W1, b1, W2, b2, noise, out);
}